// LiDAR_Loss_4784593568318
// MI455X (gfx1250) — compile-verified
//
#include <hip/hip_runtime.h>

// Problem constants (from the reference)
#define BB   8
#define HH   352
#define WW   1216
#define NL   16384           // lidar points per batch
#define NMID 1214            // valid mids per batch
#define NPAD 1216            // padded to 76 tiles of 16
#define NTILES 76            // NPAD / 16
#define MTILES 1024          // NL / 16

typedef __attribute__((ext_vector_type(2))) float v2f;
typedef __attribute__((ext_vector_type(8))) float v8f;

// ---------------------------------------------------------------------------
// Kernel 1: build mid[b][c] (3-row column average, then 3-column average),
// pad columns [1214,1216) with a huge sentinel, zero the double accumulators.
// ---------------------------------------------------------------------------
__global__ void lidar_prep_kernel(const float* __restrict__ img,
                                  float* __restrict__ mid,
                                  double* __restrict__ sums) {
    int idx = blockIdx.x * blockDim.x + threadIdx.x;
    if (idx == 0) { sums[0] = 0.0; sums[1] = 0.0; }
    if (idx >= BB * NPAD) return;
    int b = idx / NPAD;
    int c = idx - b * NPAD;
    float v;
    if (c < NMID) {
        const float* base = img + (size_t)b * HH * WW + (size_t)(HH - 3) * WW;
        // col_avg = rows.mean(axis=1); mid = 3-col mean of col_avg (same order as ref)
        float ca0 = (base[c    ] + base[WW + c    ] + base[2 * WW + c    ]) / 3.0f;
        float ca1 = (base[c + 1] + base[WW + c + 1] + base[2 * WW + c + 1]) / 3.0f;
        float ca2 = (base[c + 2] + base[WW + c + 2] + base[2 * WW + c + 2]) / 3.0f;
        v = (ca0 + ca1 + ca2) / 3.0f;
    } else {
        v = 1.0e9f;              // sentinel: (ld - 1e9)^2 ~ 1e18, never the min
    }
    mid[idx] = v;
}

// ---------------------------------------------------------------------------
// Kernel 2: cham_x via WMMA. One wave per 16-lidar-point tile (8192 waves).
// D[m][n] = ld[m]*1 + 1*(-mid[n])  (exact difference), squared per lane,
// min-reduced over n in registers + 16-lane xor shuffles.
// A 16x4 layout: VGPR0 lanes0-15 = K0, VGPR1 lanes0-15 = K1 (upper lanes = K2,K3 = 0).
// B 4x16 layout: VGPR0 lanes0-15 = K0 row, VGPR1 lanes0-15 = K1 row.
// D 16x16: VGPR r = row M=r (lanes 0-15) / M=r+8 (lanes 16-31), lane&15 = N.
// ---------------------------------------------------------------------------
__global__ void lidar_chamx_wmma_kernel(const float* __restrict__ lidar,
                                        const float* __restrict__ mid,
                                        double* __restrict__ sums) {
    const int lane = threadIdx.x & 31;
    const int wave = blockIdx.x * (blockDim.x >> 5) + (threadIdx.x >> 5);
    const int b     = wave >> 10;      // wave / MTILES
    const int mtile = wave & 1023;     // wave % MTILES
    const int m0    = mtile << 4;

    const bool lo = lane < 16;
    const float lv = lidar[(size_t)b * NL + m0 + (lane & 15)];

    v2f A;
    A.x = lo ? lv   : 0.0f;            // A[m][0] = ld[m]
    A.y = lo ? 1.0f : 0.0f;            // A[m][1] = 1

    float acc[8];
#pragma unroll
    for (int r = 0; r < 8; ++r) acc[r] = 3.0e38f;

    const float* midb = mid + (size_t)b * NPAD;
    for (int j = 0; j < NTILES; ++j) {
        float mv = midb[j * 16 + (lane & 15)];
        v2f Bf;
        Bf.x = lo ? 1.0f : 0.0f;       // B[0][n] = 1
        Bf.y = lo ? -mv  : 0.0f;       // B[1][n] = -mid[n]
        v8f C = {0.f, 0.f, 0.f, 0.f, 0.f, 0.f, 0.f, 0.f};
        v8f D = __builtin_amdgcn_wmma_f32_16x16x4_f32(
            /*neg_a=*/false, A, /*neg_b=*/false, Bf,
            /*c_mod=*/(short)0, C, /*reuse_a=*/false, /*reuse_b=*/false);
#pragma unroll
        for (int r = 0; r < 8; ++r) {
            float t = D[r];            // exact ld[m] - mid[n]
            float d = t * t;
            acc[r] = fminf(acc[r], d);
        }
    }

    // Min over the 16 lanes that share one M row (xor within 16-lane groups),
    // then sum the 8 (low half: M=0..7) + 8 (high half: M=8..15) row minima.
    float s = 0.0f;
#pragma unroll
    for (int r = 0; r < 8; ++r) {
        float v = acc[r];
        v = fminf(v, __shfl_xor(v, 1, 32));
        v = fminf(v, __shfl_xor(v, 2, 32));
        v = fminf(v, __shfl_xor(v, 4, 32));
        v = fminf(v, __shfl_xor(v, 8, 32));
        s += v;
    }
    s += __shfl_xor(s, 16, 32);        // combine the two half-wave row sets
    if (lane == 0) atomicAdd(&sums[0], (double)s);
}

// ---------------------------------------------------------------------------
// Kernel 3: cham_y. Block = 256 mids of one batch; lidar streamed through LDS
// in 8 KB tiles. 40 blocks total.
// ---------------------------------------------------------------------------
__global__ void lidar_chamy_kernel(const float* __restrict__ lidar,
                                   const float* __restrict__ mid,
                                   double* __restrict__ sums) {
    __shared__ float tile[2048];
    const int b     = blockIdx.x / 5;
    const int chunk = blockIdx.x - b * 5;
    const int tid   = threadIdx.x;
    const int n     = chunk * 256 + tid;
    const bool valid = n < NMID;
    const float mv = valid ? mid[(size_t)b * NPAD + n] : 0.0f;

    float minv = 3.0e38f;
    const float* lb = lidar + (size_t)b * NL;
    for (int t = 0; t < NL; t += 2048) {
#pragma unroll
        for (int k = 0; k < 2048; k += 256) tile[k + tid] = lb[t + k + tid];
        __syncthreads();
        for (int k = 0; k < 2048; ++k) {
            float diff = tile[k] - mv;
            minv = fminf(minv, diff * diff);
        }
        __syncthreads();
    }

    float s = valid ? minv : 0.0f;
    s += __shfl_xor(s, 1, 32);
    s += __shfl_xor(s, 2, 32);
    s += __shfl_xor(s, 4, 32);
    s += __shfl_xor(s, 8, 32);
    s += __shfl_xor(s, 16, 32);
    if ((tid & 31) == 0) atomicAdd(&sums[1], (double)s);
}

// ---------------------------------------------------------------------------
// Kernel 4: finalize the scalar.
// ---------------------------------------------------------------------------
__global__ void lidar_finalize_kernel(const double* __restrict__ sums,
                                      float* __restrict__ out) {
    if (blockIdx.x == 0 && threadIdx.x == 0) {
        out[0] = (float)(sums[0] / (double)(BB * NL) +
                         sums[1] / (double)(BB * NMID));
    }
}

// ---------------------------------------------------------------------------
// d_ws layout: [0,16) two doubles (sumx, sumy); [16, 16+8*1216*4) mid floats.
// ---------------------------------------------------------------------------
extern "C" void kernel_launch(void* const* d_in, const int* in_sizes, int n_in,
                              void* d_out, int out_size, void* d_ws, size_t ws_size,
                              hipStream_t stream) {
    const float* img   = (const float*)d_in[0];   // output: (8,1,352,1216) f32
    const float* lidar = (const float*)d_in[1];   // lidar : (8,1,16384)   f32
    double* sums = (double*)d_ws;
    float*  mid  = (float*)((char*)d_ws + 16);
    float*  out  = (float*)d_out;

    lidar_prep_kernel<<<(BB * NPAD + 255) / 256, 256, 0, stream>>>(img, mid, sums);
    lidar_chamx_wmma_kernel<<<(BB * MTILES) / 8, 256, 0, stream>>>(lidar, mid, sums);
    lidar_chamy_kernel<<<BB * 5, 256, 0, stream>>>(lidar, mid, sums);
    lidar_finalize_kernel<<<1, 1, 0, stream>>>(sums, out);
}